// GATMolGCN_48962627175094
// MI455X (gfx1250) — compile-verified
//
#include <hip/hip_runtime.h>

typedef __attribute__((ext_vector_type(2))) float v2f;
typedef __attribute__((ext_vector_type(8))) float v8f;

#define IN_C 128
#define HID 64
#define HEADS 4
#define F1 (HEADS * HID) /* 256 */
#define OUT_C 64
#define NEG_SLOPE 0.2f

// ---------------- float atomic max/min via monotonic int mapping ----------
// Safe for mixed-sign values given +/-inf initialization:
//  - positives order identically as signed ints
//  - negatives order inversely as unsigned ints
__device__ __forceinline__ void atomicMaxF(float* addr, float v) {
    if (v >= 0.0f) atomicMax((int*)addr, __float_as_int(v));
    else           atomicMin((unsigned int*)addr, __float_as_uint(v));
}
__device__ __forceinline__ void atomicMinF(float* addr, float v) {
    if (v >= 0.0f) atomicMin((int*)addr, __float_as_int(v));
    else           atomicMax((unsigned int*)addr, __float_as_uint(v));
}

// ---------------- generic fill ----------------
__global__ void k_fill(float* __restrict__ p, long long n, float v) {
    long long i = (long long)blockIdx.x * blockDim.x + threadIdx.x;
    if (i < n) p[i] = v;
}

// ---------------- fp32 WMMA GEMM: C[M,N] = A[M,K] * B[K,N], row-major -----
// One wave per 16x16 output tile; V_WMMA_F32_16X16X4_F32, K stepped by 4.
// Requires M%16==0, N%16==0, K%4==0 (holds: M=50000, K in {128,256}, N in {256,64}).
// A-layout (ISA 7.12.2): lane holds M=lane&15, K pair 2*(lane>>4)+{0,1} -> float2 load.
// B mirrored (N in lanes); C/D: m = v + 8*(lane>>4), n = lane&15.
__global__ __launch_bounds__(256) void k_gemm_wmma(const float* __restrict__ A,
                                                   const float* __restrict__ B,
                                                   float* __restrict__ C,
                                                   int M, int K, int N) {
    const int lane   = threadIdx.x & 31;
    const int wave   = threadIdx.x >> 5;
    const int tilesN = N >> 4;
    const int nTiles = (M >> 4) * tilesN;
    const int tile   = blockIdx.x * (blockDim.x >> 5) + wave;  // wave-uniform
    if (tile >= nTiles) return;                                 // whole-wave exit, EXEC stays all-1s

    const int tm    = tile / tilesN;
    const int tn    = tile % tilesN;
    const int row   = tm * 16 + (lane & 15);       // A row for this lane
    const int col   = tn * 16 + (lane & 15);       // B/C column for this lane
    const int khalf = (lane >> 4) << 1;            // 0 or 2

    v8f acc = {};
    for (int k = 0; k < K; k += 4) {
        const int k0 = k + khalf;                  // even -> 8B aligned
        v2f a = *(const v2f*)(A + (size_t)row * K + k0);
        v2f b;
        b.x = B[(size_t)k0 * N + col];
        b.y = B[(size_t)(k0 + 1) * N + col];
        acc = __builtin_amdgcn_wmma_f32_16x16x4_f32(
            false, a, false, b, (short)0, acc, false, false);
    }
    const int rbase = tm * 16 + ((lane >> 4) << 3);
    for (int v = 0; v < 8; ++v)
        C[(size_t)(rbase + v) * N + col] = acc[v];
}

// ---------------- per-(node,head) attention scores ------------------------
__global__ void k_scores(const float* __restrict__ h, const float* __restrict__ a_src,
                         const float* __restrict__ a_dst, float* __restrict__ s_src,
                         float* __restrict__ s_dst, int n, int H, int D) {
    int i = blockIdx.x * blockDim.x + threadIdx.x;  // node*H + head
    if (i >= n * H) return;
    int node = i / H, head = i % H;
    const float* hp = h + (size_t)node * H * D + (size_t)head * D;
    const float* as = a_src + head * D;
    const float* ad = a_dst + head * D;
    float ss = 0.0f, sd = 0.0f;
    for (int d = 0; d < D; ++d) { float x = hp[d]; ss += x * as[d]; sd += x * ad[d]; }
    s_src[i] = ss;
    s_dst[i] = sd;
}

// edge e < E : from edge_index; e >= E : self-loop (e - E)
__device__ __forceinline__ void edge_ends(const int* __restrict__ ei, int E, int e,
                                          int& s, int& d) {
    if (e < E) { s = ei[e]; d = ei[E + e]; }
    else       { s = e - E; d = e - E; }
}

__device__ __forceinline__ float leaky(float x) {
    return x > 0.0f ? x : NEG_SLOPE * x;
}

// ---------------- segment-softmax pass 1: per-dst max ---------------------
__global__ void k_edge_max(const int* __restrict__ ei, int E, int Etot, int H,
                           const float* __restrict__ s_src, const float* __restrict__ s_dst,
                           float* __restrict__ m) {
    long long i = (long long)blockIdx.x * blockDim.x + threadIdx.x;
    if (i >= (long long)Etot * H) return;
    int e = (int)(i / H), hd = (int)(i % H);
    int s, d; edge_ends(ei, E, e, s, d);
    float x = leaky(s_src[s * H + hd] + s_dst[d * H + hd]);
    atomicMaxF(&m[d * H + hd], x);
}

// ---------------- pass 2: per-dst sum of exp ------------------------------
__global__ void k_edge_sum(const int* __restrict__ ei, int E, int Etot, int H,
                           const float* __restrict__ s_src, const float* __restrict__ s_dst,
                           const float* __restrict__ m, float* __restrict__ denom) {
    long long i = (long long)blockIdx.x * blockDim.x + threadIdx.x;
    if (i >= (long long)Etot * H) return;
    int e = (int)(i / H), hd = (int)(i % H);
    int s, d; edge_ends(ei, E, e, s, d);
    float x = leaky(s_src[s * H + hd] + s_dst[d * H + hd]);
    atomicAdd(&denom[d * H + hd], expf(x - m[d * H + hd]));
}

// ---------------- pass 2.5: per-(edge,head) softmax weight ----------------
__global__ void k_edge_alpha(const int* __restrict__ ei, int E, int Etot, int H,
                             const float* __restrict__ s_src, const float* __restrict__ s_dst,
                             const float* __restrict__ m, const float* __restrict__ denom,
                             float* __restrict__ alpha) {
    long long i = (long long)blockIdx.x * blockDim.x + threadIdx.x;
    if (i >= (long long)Etot * H) return;
    int e = (int)(i / H), hd = (int)(i % H);
    int s, d; edge_ends(ei, E, e, s, d);
    float x = leaky(s_src[s * H + hd] + s_dst[d * H + hd]);
    alpha[i] = expf(x - m[d * H + hd]) / denom[d * H + hd];
}

// ---------------- pass 3: weighted gather + scatter-add -------------------
// Thread per (edge, feature): one alpha broadcast load + one coalesced
// feature gather + one GLOBAL_ATOMIC_ADD_F32 scatter.
__global__ void k_edge_agg(const int* __restrict__ ei, int E, int Etot, int H, int D,
                           const float* __restrict__ alpha,
                           const float* __restrict__ hfeat, float* __restrict__ agg) {
    long long i = (long long)blockIdx.x * blockDim.x + threadIdx.x;  // Etot*H*D
    const int Ff = H * D;
    if (i >= (long long)Etot * Ff) return;
    int f  = (int)(i % Ff);
    int e  = (int)(i / Ff);
    int hd = f / D;
    int s, d; edge_ends(ei, E, e, s, d);
    float a = alpha[(size_t)e * H + hd];
    atomicAdd(&agg[(size_t)d * Ff + f], hfeat[(size_t)s * Ff + f] * a);
}

// ---------------- bias + ELU (in place) -----------------------------------
__global__ void k_bias_elu(float* __restrict__ agg, const float* __restrict__ b,
                           long long n, int Ff) {
    long long i = (long long)blockIdx.x * blockDim.x + threadIdx.x;
    if (i >= n * Ff) return;
    float x = agg[i] + b[(int)(i % Ff)];
    agg[i] = x > 0.0f ? x : (expf(x) - 1.0f);
}

// ---------------- final: out[f] = min_n (agg2[n,f] + b2[f]) ---------------
__global__ void k_min_reduce(const float* __restrict__ agg2, const float* __restrict__ b2,
                             float* __restrict__ out, int n) {
    const int f    = threadIdx.x;        // 64 threads
    const int base = blockIdx.x * 256;   // 256 nodes per block
    float bias  = b2[f];
    float local = __builtin_inff();
    for (int i = 0; i < 256; ++i) {
        int node = base + i;
        if (node < n) {
            float x = agg2[(size_t)node * OUT_C + f] + bias;
            local = fminf(local, x);
        }
    }
    atomicMinF(&out[f], local);
}

static inline int cdiv_ll(long long a, long long b) { return (int)((a + b - 1) / b); }

extern "C" void kernel_launch(void* const* d_in, const int* in_sizes, int n_in,
                              void* d_out, int out_size, void* d_ws, size_t ws_size,
                              hipStream_t stream) {
    const float* x    = (const float*)d_in[0];
    const int*   ei   = (const int*)d_in[1];
    const float* W1   = (const float*)d_in[2];
    const float* a_s1 = (const float*)d_in[3];
    const float* a_d1 = (const float*)d_in[4];
    const float* b1   = (const float*)d_in[5];
    const float* W2   = (const float*)d_in[6];
    const float* a_s2 = (const float*)d_in[7];
    const float* a_d2 = (const float*)d_in[8];
    const float* b2   = (const float*)d_in[9];
    float* out = (float*)d_out;

    const int n    = in_sizes[0] / IN_C;  // 50000
    const int E    = in_sizes[1] / 2;     // 800000
    const int Etot = E + n;               // edges + self-loops

    // ---- workspace carve-up (floats) ----
    float* ws = (float*)d_ws;
    size_t o = 0;
    float* h1    = ws + o; o += (size_t)n * F1;       // GEMM1 output
    float* agg1  = ws + o; o += (size_t)n * F1;       // layer-1 aggregate -> ELU'd features
    float* h2    = ws + o; o += (size_t)n * OUT_C;    // GEMM2 output
    float* agg2  = ws + o; o += (size_t)n * OUT_C;    // layer-2 aggregate
    float* alph  = ws + o; o += (size_t)Etot * HEADS; // per-(edge,head) alpha (reused by layer 2)
    float* ssrc1 = ws + o; o += (size_t)n * HEADS;
    float* sdst1 = ws + o; o += (size_t)n * HEADS;
    float* m1    = ws + o; o += (size_t)n * HEADS;
    float* den1  = ws + o; o += (size_t)n * HEADS;
    float* ssrc2 = ws + o; o += (size_t)n;
    float* sdst2 = ws + o; o += (size_t)n;
    float* m2    = ws + o; o += (size_t)n;
    float* den2  = ws + o; o += (size_t)n;
    (void)ws_size; (void)n_in; (void)out_size;

    const float NINF = -__builtin_inff();
    const float PINF =  __builtin_inff();

    // ---- init (every call: ws/out are not re-poisoned between replays) ----
    k_fill<<<cdiv_ll((long long)n * F1, 256), 256, 0, stream>>>(agg1, (long long)n * F1, 0.0f);
    k_fill<<<cdiv_ll((long long)n * OUT_C, 256), 256, 0, stream>>>(agg2, (long long)n * OUT_C, 0.0f);
    k_fill<<<cdiv_ll((long long)n * HEADS, 256), 256, 0, stream>>>(m1, (long long)n * HEADS, NINF);
    k_fill<<<cdiv_ll((long long)n * HEADS, 256), 256, 0, stream>>>(den1, (long long)n * HEADS, 0.0f);
    k_fill<<<cdiv_ll(n, 256), 256, 0, stream>>>(m2, n, NINF);
    k_fill<<<cdiv_ll(n, 256), 256, 0, stream>>>(den2, n, 0.0f);
    k_fill<<<1, 64, 0, stream>>>(out, OUT_C, PINF);

    // ---- layer 1 ----
    {
        int tiles = (n / 16) * (F1 / 16);
        k_gemm_wmma<<<cdiv_ll(tiles, 8), 256, 0, stream>>>(x, W1, h1, n, IN_C, F1);
    }
    k_scores<<<cdiv_ll((long long)n * HEADS, 256), 256, 0, stream>>>(h1, a_s1, a_d1, ssrc1, sdst1, n, HEADS, HID);
    k_edge_max<<<cdiv_ll((long long)Etot * HEADS, 256), 256, 0, stream>>>(ei, E, Etot, HEADS, ssrc1, sdst1, m1);
    k_edge_sum<<<cdiv_ll((long long)Etot * HEADS, 256), 256, 0, stream>>>(ei, E, Etot, HEADS, ssrc1, sdst1, m1, den1);
    k_edge_alpha<<<cdiv_ll((long long)Etot * HEADS, 256), 256, 0, stream>>>(ei, E, Etot, HEADS, ssrc1, sdst1, m1, den1, alph);
    k_edge_agg<<<cdiv_ll((long long)Etot * F1, 256), 256, 0, stream>>>(ei, E, Etot, HEADS, HID, alph, h1, agg1);
    k_bias_elu<<<cdiv_ll((long long)n * F1, 256), 256, 0, stream>>>(agg1, b1, n, F1);

    // ---- layer 2 (heads=1, D=OUT_C) ----
    {
        int tiles = (n / 16) * (OUT_C / 16);
        k_gemm_wmma<<<cdiv_ll(tiles, 8), 256, 0, stream>>>(agg1, W2, h2, n, F1, OUT_C);
    }
    k_scores<<<cdiv_ll(n, 256), 256, 0, stream>>>(h2, a_s2, a_d2, ssrc2, sdst2, n, 1, OUT_C);
    k_edge_max<<<cdiv_ll(Etot, 256), 256, 0, stream>>>(ei, E, Etot, 1, ssrc2, sdst2, m2);
    k_edge_sum<<<cdiv_ll(Etot, 256), 256, 0, stream>>>(ei, E, Etot, 1, ssrc2, sdst2, m2, den2);
    k_edge_alpha<<<cdiv_ll(Etot, 256), 256, 0, stream>>>(ei, E, Etot, 1, ssrc2, sdst2, m2, den2, alph);
    k_edge_agg<<<cdiv_ll((long long)Etot * OUT_C, 256), 256, 0, stream>>>(ei, E, Etot, 1, OUT_C, alph, h2, agg2);

    // ---- bias + column-wise min over all nodes ----
    k_min_reduce<<<cdiv_ll(n, 256), 64, 0, stream>>>(agg2, b2, out, n);
}